// Bbox_loss_44040594653650
// MI455X (gfx1250) — compile-verified
//
#include <hip/hip_runtime.h>

typedef __attribute__((ext_vector_type(2))) float v2f;
typedef __attribute__((ext_vector_type(8))) float v8f;

#define NLEV   3
#define BATCH  4
#define MPER   128
#define APV    3
#define TOTAL  (NLEV * BATCH * MPER)   // 1536 gather items
#define NTHREADS 1024                  // 32 waves, one WGP

// Dual-quantity wave32 reduction via V_WMMA_F32_16X16X4_F32.
// A (16x4 f32): lanes 0-15 -> VGPR0=K0, VGPR1=K1 ; lanes 16-31 -> VGPR0=K2, VGPR1=K3
//   => A[m][0]=loss(lane m), A[m][1]=wt(lane m), A[m][2]=loss(lane m+16), A[m][3]=wt(lane m+16)
// B (4x16 f32): VGPR0 = rows K0|K1, VGPR1 = rows K2|K3, striped across lanes.
//   Set lane0 (K0,N0 / K2,N0) and lane17 (K1,N1 / K3,N1) to 1.0:
//   => D[m][0] = loss(m)+loss(m+16), D[m][1] = wt(m)+wt(m+16), other cols 0.
// D (16x16 f32): lane0 VGPRs0..7 = D[0..7][0], lane16 = D[8..15][0],
//                lane1 = D[0..7][1],           lane17 = D[8..15][1].
__device__ __forceinline__ void wave_reduce2_wmma(float loss, float wt,
                                                  float& loss_out, float& wt_out) {
    const int lane = threadIdx.x & 31;
    v2f a; a.x = loss; a.y = wt;
    const float bsel = (lane == 0 || lane == 17) ? 1.0f : 0.0f;
    v2f b; b.x = bsel; b.y = bsel;
    v8f c = {};
    v8f d = __builtin_amdgcn_wmma_f32_16x16x4_f32(
        /*neg_a=*/false, a, /*neg_b=*/false, b,
        /*c_mod=*/(short)0, c, /*reuse_a=*/false, /*reuse_b=*/false);
    float s = d[0] + d[1] + d[2] + d[3] + d[4] + d[5] + d[6] + d[7];
    loss_out = __shfl(s, 0, 32)  + __shfl(s, 16, 32);
    wt_out   = __shfl(s, 1, 32)  + __shfl(s, 17, 32);
}

__global__ __launch_bounds__(NTHREADS, 1)
void bbox_loss_kernel(const float* __restrict__ pred0, const int* __restrict__ coord0,
                      const float* __restrict__ diff0,
                      const float* __restrict__ pred1, const int* __restrict__ coord1,
                      const float* __restrict__ diff1,
                      const float* __restrict__ pred2, const int* __restrict__ coord2,
                      const float* __restrict__ diff2,
                      float* __restrict__ out) {
    __shared__ float lds_loss[32];
    __shared__ float lds_wt[32];

    const float* preds[NLEV]  = {pred0, pred1, pred2};
    const int*   coords[NLEV] = {coord0, coord1, coord2};
    const float* diffs[NLEV]  = {diff0, diff1, diff2};
    const int    S[NLEV]      = {96, 48, 24};

    const int tid = threadIdx.x;

    // Warm the small dense side tables (global_prefetch_b8).
    if (tid < NLEV) {
        __builtin_prefetch(coords[tid], 0, 0);
        __builtin_prefetch(diffs[tid], 0, 0);
    }

    float loss = 0.0f, wt = 0.0f;

    // 1536 items over 1024 threads: per-wave-uniform trip counts (boundary at tid 512).
    for (int item = tid; item < TOTAL; item += NTHREADS) {
        const int lvl = item / (BATCH * MPER);
        const int rem = item - lvl * (BATCH * MPER);
        const int b   = rem / MPER;
        const int m   = rem - b * MPER;

        const int s  = S[lvl];
        const int s3 = s * s * s;

        const int* cptr = coords[lvl] + ((b * MPER + m) << 2);
        const int c0 = cptr[0], c1 = cptr[1], c2 = cptr[2], c3 = cptr[3];
        const float fm = (c0 > -1) ? 1.0f : 0.0f;   // valid-anchor mask (raw coord)

        const int a  = max(c0, 0);
        const int dz = max(c1, 0);
        const int hy = max(c2, 0);
        const int wx = max(c3, 0);

        const int spatial = (dz * s + hy) * s + wx;
        const int base    = (b * (6 * APV) + a) * s3 + spatial;   // channel = j*A + a

        const float* pp   = preds[lvl];
        const float* dptr = diffs[lvl] + (b * MPER + m) * 6;

        float lsum = 0.0f;
        #pragma unroll
        for (int j = 0; j < 6; ++j) {
            const float g  = pp[base + j * APV * s3];
            const float df = dptr[j];
            const float dd = g - df;
            const float ad = fabsf(dd);
            lsum += (ad < 1.0f) ? (0.5f * dd * dd) : (ad - 0.5f);
        }
        loss += fm * lsum;   // masked smooth-L1 sum
        wt   += fm;          // one count per valid (b,m), matching jnp.sum(mask)
    }

    // Stage 1: per-wave reduction via WMMA (EXEC all-1s here).
    float wl, ww;
    wave_reduce2_wmma(loss, wt, wl, ww);

    const int wave = tid >> 5;
    const int lane = tid & 31;
    if (lane == 0) { lds_loss[wave] = wl; lds_wt[wave] = ww; }
    __syncthreads();

    // Stage 2: wave 0 reduces the 32 wave partials with one more WMMA.
    if (wave == 0) {
        float tl, tw;
        wave_reduce2_wmma(lds_loss[lane], lds_wt[lane], tl, tw);
        if (lane == 0) {
            out[0] = tl;   // reg_loss
            out[1] = tw;   // reg_weight
        }
    }
}

extern "C" void kernel_launch(void* const* d_in, const int* in_sizes, int n_in,
                              void* d_out, int out_size, void* d_ws, size_t ws_size,
                              hipStream_t stream) {
    // setup_inputs() dict order: (pred, coord, diff) per level 0..2
    const float* pred0  = (const float*)d_in[0];
    const int*   coord0 = (const int*)  d_in[1];
    const float* diff0  = (const float*)d_in[2];
    const float* pred1  = (const float*)d_in[3];
    const int*   coord1 = (const int*)  d_in[4];
    const float* diff1  = (const float*)d_in[5];
    const float* pred2  = (const float*)d_in[6];
    const int*   coord2 = (const int*)  d_in[7];
    const float* diff2  = (const float*)d_in[8];
    float* out = (float*)d_out;        // [loss, weight]

    bbox_loss_kernel<<<1, NTHREADS, 0, stream>>>(
        pred0, coord0, diff0, pred1, coord1, diff1, pred2, coord2, diff2, out);
}